// StarSpace_42056319763043
// MI455X (gfx1250) — compile-verified
//
#include <hip/hip_runtime.h>
#include <math.h>

// ---------------------------------------------------------------------------
// StarSpace embedding-bag forward for MI455X (gfx1250).
//
// Memory-bound gather workload: 262 MB of random 256B row reads from a
// 25.6 MB table that lives in the 192 MB L2. No matmul structure -> WMMA is
// not applicable; the matching CDNA5 path is the Tensor Data Mover in
// gather mode (TENSOR_LOAD_TO_LDS, 32-bit row indices, 8 rows / 2KB per op),
// pipelined 4 deep with s_wait_tensorcnt.
// ---------------------------------------------------------------------------

#define V_ROWS  100000
#define DIM     64
#define BATCH   4096
#define SEQ     50
#define KNEG    3
#define NBAGS   (BATCH * (2 + KNEG))   // 20480 output rows, out offset = bag*DIM
#define MAXNORM 20.0f

typedef unsigned int v4u __attribute__((ext_vector_type(4)));
typedef int          v8i __attribute__((ext_vector_type(8)));
typedef int          v4i __attribute__((ext_vector_type(4)));

#if defined(__HIP_DEVICE_COMPILE__) && \
    __has_builtin(__builtin_amdgcn_tensor_load_to_lds) && \
    __has_builtin(__builtin_amdgcn_s_wait_tensorcnt)
#define USE_TDM 1
#else
#define USE_TDM 0
#endif

#define MEMBAR() asm volatile("" ::: "memory")

// ---------------------------------------------------------------------------
// Kernel 1: per-row renorm scale:  scale[r] = norm>20 ? 20/norm : 1
// one wave32 per row; lanes load float2 -> shfl_xor tree reduce.
// ---------------------------------------------------------------------------
__global__ __launch_bounds__(256) void scales_kernel(const float* __restrict__ emb,
                                                     float* __restrict__ scl) {
  int row  = blockIdx.x * 8 + (threadIdx.x >> 5);
  int lane = threadIdx.x & 31;
  if (row >= V_ROWS) return;
  const float2* r = (const float2*)(emb + row * DIM);
  float2 v = r[lane];
  float s = v.x * v.x + v.y * v.y;
#pragma unroll
  for (int off = 16; off > 0; off >>= 1) s += __shfl_xor(s, off, 32);
  if (lane == 0) {
    float n = sqrtf(s);
    scl[row] = (n > MAXNORM) ? (MAXNORM / n) : 1.0f;
  }
}

#if USE_TDM
// ---------------------------------------------------------------------------
// Issue one TDM gather: 8 embedding rows (32-bit indices) -> LDS.
// D# built per CDNA5 ISA (gather mode).
// This toolchain's builtin is the 6-arg form:
//   (uint32x4 g0, int32x8 g1, int32x4 g2, int32x4 g3, int32x8 extra, i32 cpol)
// The D# defines only 4 groups (128b+256b+128b+128b); the trailing int32x8 is
// zero-filled (probe-verified to compile with zeros).
// ---------------------------------------------------------------------------
__device__ __forceinline__ void tdm_issue(const float* __restrict__ emb,
                                          unsigned lds_byte,
                                          const int* __restrict__ tokp,
                                          int cbase, int nvalid) {
  unsigned long long ga = (unsigned long long)(uintptr_t)emb;
  v4u g0;
  g0[0] = 0xC0000001u;                    // count=1 | gather_index_size=32b | gather_mode=1
  g0[1] = lds_byte;                       // lds_addr (bytes)
  g0[2] = (unsigned)(ga & 0xFFFFFFFFull); // global_addr[31:0]
  g0[3] = (unsigned)((ga >> 32) & 0x01FFFFFFull) | (2u << 30); // addr[56:32] | type=2

  v8i g1;
  g1[0] = (int)0x00020000u;                             // data_size=2 (4B elements)
  g1[1] = (int)(((unsigned)DIM & 0xFFFFu) << 16);       // tensor_dim0 lo16 (=64)
  g1[2] = (int)(((unsigned)V_ROWS & 0xFFFFu) << 16);    // dim0 hi16=0 | tensor_dim1 lo16
  g1[3] = (int)(((unsigned)V_ROWS >> 16) |
                ((unsigned)DIM << 16));                 // tensor_dim1 hi16 | tile_dim0=64
  g1[4] = nvalid;                                       // tile_dim1 = #valid indices
  g1[5] = DIM;                                          // tensor_dim0_stride lo32 = 64
  g1[6] = 0;
  g1[7] = 0;

  int t[8];
#pragma unroll
  for (int j = 0; j < 8; ++j) t[j] = (j < nvalid) ? tokp[cbase + j] : 0;
  v4i g2 = {t[0], t[1], t[2], t[3]};
  v4i g3 = {t[4], t[5], t[6], t[7]};
  v8i g4 = {0, 0, 0, 0, 0, 0, 0, 0};                    // beyond-D# extra group: zeros

  MEMBAR();
  __builtin_amdgcn_tensor_load_to_lds(g0, g1, g2, g3, g4, 0);
  MEMBAR();
}
#endif

// Accumulate n (<=8) scaled rows of a gathered chunk. lbuf = LDS chunk base.
__device__ __forceinline__ void acc_chunk(const float* __restrict__ lbuf,
                                          const float* __restrict__ scl,
                                          const int* __restrict__ tokp,
                                          int cbase, int n, int lane,
                                          float& ax, float& ay) {
#pragma unroll
  for (int j = 0; j < 8; ++j) {
    if (j < n) {
      int   t  = tokp[cbase + j];   // uniform -> scalar load
      float sc = scl[t];            // uniform -> scalar load
      const float2* row = (const float2*)(lbuf + j * DIM);
      float2 v = row[lane];         // ds_load_b64, conflict-free
      ax += sc * v.x;
      ay += sc * v.y;
    }
  }
}

// ---------------------------------------------------------------------------
// Kernel 2: one wave32 per bag. 50 tokens = 6x8 + 2 gather chunks.
// 4 LDS buffers (8KB/WG), TDM pipeline depth 4, literal tensorcnt waits.
// ---------------------------------------------------------------------------
__global__ __launch_bounds__(32) void bag_kernel(const float* __restrict__ emb,
                                                 const float* __restrict__ scl,
                                                 const int* __restrict__ tl,
                                                 const int* __restrict__ tr,
                                                 const int* __restrict__ tn,
                                                 float* __restrict__ out) {
  __shared__ float lbuf[4][8 * DIM];    // 4 x 2KB quad buffer
  const int bag  = blockIdx.x;
  const int lane = threadIdx.x;

  const int* tokp;
  if (bag < BATCH)          tokp = tl + bag * SEQ;
  else if (bag < 2 * BATCH) tokp = tr + (bag - BATCH) * SEQ;
  else                      tokp = tn + (bag - 2 * BATCH) * SEQ;

  float ax = 0.0f, ay = 0.0f;

#if USE_TDM
  const unsigned lb = (unsigned)(uintptr_t)(&lbuf[0][0]);

  tdm_issue(emb, lb + 0 * 2048u, tokp,  0, 8);
  tdm_issue(emb, lb + 1 * 2048u, tokp,  8, 8);
  tdm_issue(emb, lb + 2 * 2048u, tokp, 16, 8);
  tdm_issue(emb, lb + 3 * 2048u, tokp, 24, 8);

  __builtin_amdgcn_s_wait_tensorcnt(3); MEMBAR();   // chunk 0 done
  acc_chunk(lbuf[0], scl, tokp,  0, 8, lane, ax, ay);
  tdm_issue(emb, lb + 0 * 2048u, tokp, 32, 8);      // chunk 4 -> buf0

  __builtin_amdgcn_s_wait_tensorcnt(3); MEMBAR();   // chunk 1 done
  acc_chunk(lbuf[1], scl, tokp,  8, 8, lane, ax, ay);
  tdm_issue(emb, lb + 1 * 2048u, tokp, 40, 8);      // chunk 5 -> buf1

  __builtin_amdgcn_s_wait_tensorcnt(3); MEMBAR();   // chunk 2 done
  acc_chunk(lbuf[2], scl, tokp, 16, 8, lane, ax, ay);
  tdm_issue(emb, lb + 2 * 2048u, tokp, 48, 2);      // chunk 6 (2 rows) -> buf2

  __builtin_amdgcn_s_wait_tensorcnt(3); MEMBAR();   // chunk 3 done
  acc_chunk(lbuf[3], scl, tokp, 24, 8, lane, ax, ay);

  __builtin_amdgcn_s_wait_tensorcnt(2); MEMBAR();   // chunk 4 done
  acc_chunk(lbuf[0], scl, tokp, 32, 8, lane, ax, ay);

  __builtin_amdgcn_s_wait_tensorcnt(1); MEMBAR();   // chunk 5 done
  acc_chunk(lbuf[1], scl, tokp, 40, 8, lane, ax, ay);

  __builtin_amdgcn_s_wait_tensorcnt(0); MEMBAR();   // chunk 6 done
  acc_chunk(lbuf[2], scl, tokp, 48, 2, lane, ax, ay);
#else
  // Fallback: direct gather (L2-resident rows), prefetch 4 tokens ahead.
  (void)lbuf;
#pragma unroll 5
  for (int j = 0; j < SEQ; ++j) {
    if (j + 4 < SEQ) {
      int tp = tokp[j + 4];
      __builtin_prefetch(emb + tp * DIM, 0, 0);   // global_prefetch_b8
    }
    int   t  = tokp[j];
    float sc = scl[t];
    const float2* row = (const float2*)(emb + t * DIM);
    float2 v = row[lane];
    ax += sc * v.x;
    ay += sc * v.y;
  }
#endif

  float2* o = (float2*)(out + bag * DIM);           // out offset == bag*64 for l|r|neg
  o[lane] = make_float2(ax, ay);
}

// ---------------------------------------------------------------------------
// Launcher. inputs: [embeddings f32 VxD, tokens_l i32 BxL, tokens_r i32 BxL,
// tokens_neg i32 BxKxL]; out: 20480x64 f32; ws: 400KB of row scales.
// ---------------------------------------------------------------------------
extern "C" void kernel_launch(void* const* d_in, const int* in_sizes, int n_in,
                              void* d_out, int out_size, void* d_ws, size_t ws_size,
                              hipStream_t stream) {
  (void)in_sizes; (void)n_in; (void)out_size; (void)ws_size;
  const float* emb = (const float*)d_in[0];
  const int*   tl  = (const int*)d_in[1];
  const int*   tr  = (const int*)d_in[2];
  const int*   tn  = (const int*)d_in[3];
  float*       out = (float*)d_out;
  float*       scl = (float*)d_ws;   // V_ROWS floats = 400 KB

  scales_kernel<<<(V_ROWS + 7) / 8, 256, 0, stream>>>(emb, scl);
  bag_kernel<<<NBAGS, 32, 0, stream>>>(emb, scl, tl, tr, tn, out);
}